// AABFM_77790447665517
// MI455X (gfx1250) — compile-verified
//
#include <hip/hip_runtime.h>
#include <hip/hip_bf16.h>
#include <stdint.h>

// Problem constants (reference: DIM=256, bs=16, H=W=64)
#define DIMC   256
#define NTOK   4096
#define BATCH  16
#define TILE_T 64

typedef float v2f __attribute__((ext_vector_type(2)));
typedef float v8f __attribute__((ext_vector_type(8)));
typedef unsigned int v4u __attribute__((ext_vector_type(4)));
typedef int   v8i __attribute__((ext_vector_type(8)));
typedef int   v4i __attribute__((ext_vector_type(4)));

// Workspace layout (floats):
//   [0,              2*4096)  : s[branch][b*256+d]   channel sums
//   [2*4096,         4*4096)  : v[branch][b*256+d]   projected vectors
//   [4*4096,  4*4096+32)      : c[branch][b]         bias dot terms
// total ~66 KB

// ------------------- kernel 1: per-(batch,channel) spatial sums -------------
__global__ __launch_bounds__(256) void k1_sums(const float* __restrict__ rgb,
                                               const float* __restrict__ evt,
                                               float* __restrict__ ws) {
  const int wave = threadIdx.x >> 5;
  const int lane = threadIdx.x & 31;
  const int row  = blockIdx.x * 8 + wave;                 // 0..8191 (rgb rows, then evt rows)
  const float* src = (row < BATCH * DIMC) ? rgb : evt;
  const float4* p  = (const float4*)(src + (size_t)(row & (BATCH * DIMC - 1)) * NTOK);
  float acc = 0.f;
  for (int i = lane; i < NTOK / 4; i += 32) {             // 32 float4 per lane
    float4 v = p[i];
    acc += (v.x + v.y) + (v.z + v.w);
  }
  for (int off = 16; off > 0; off >>= 1) acc += __shfl_xor(acc, off, 32);
  if (lane == 0) ws[row] = acc;
}

// ------------------- kernel 2: batched projections via f32 WMMA -------------
// KS(256x16) = Wk * S(256x16) + 4096*bk ;  V(256x16) = Wq^T * KS ;  c = bq . KS
// WMMA 16x16x4 f32 fragment layout (ISA 7.12.2):
//   A (16x4): lanes 0-15 -> M=lane, vgpr{0,1}=K{0,1}; lanes 16-31 -> K{2,3}
//   B (4x16): lanes 0-15 -> N=lane, vgpr{0,1}=K{0,1}; lanes 16-31 -> K{2,3}
//   C/D     : vgpr r -> row r (lanes 0-15) / row r+8 (lanes 16-31), col = lane&15
__global__ __launch_bounds__(256) void k2_proj_wmma(
    const float* __restrict__ Wq_a, const float* __restrict__ bq_a,
    const float* __restrict__ Wk_a, const float* __restrict__ bk_a,
    const float* __restrict__ Wq_d, const float* __restrict__ bq_d,
    const float* __restrict__ Wk_d, const float* __restrict__ bk_d,
    float* __restrict__ ws) {
  const int branch = blockIdx.x;
  const float* Wq = branch ? Wq_d : Wq_a;
  const float* bq = branch ? bq_d : bq_a;
  const float* Wk = branch ? Wk_d : Wk_a;
  const float* bk = branch ? bk_d : bk_a;
  const float* sums = ws + branch * (BATCH * DIMC);
  float* vout = ws + 2 * BATCH * DIMC + branch * (BATCH * DIMC);
  float* cout = ws + 4 * BATCH * DIMC + branch * BATCH;

  __shared__ float S_lds[DIMC * 16];   // S[d][b]
  __shared__ float KS_lds[DIMC * 16];  // KS[e][b]

  for (int idx = threadIdx.x; idx < DIMC * 16; idx += 256) {
    int b = idx & 15, d = idx >> 4;
    S_lds[d * 16 + b] = sums[b * DIMC + d];
  }
  __syncthreads();

  const int wave   = threadIdx.x >> 5;
  const int lane   = threadIdx.x & 31;
  const int nn     = lane & 15;            // M for A-frag, N for B/C/D frags
  const int kh     = (lane >> 4) << 1;     // K sub-offset: 0 or 2
  const int rowoff = (lane >> 4) << 3;     // C/D row offset: 0 or 8

  // Phase 1: KS = Wk * S + 4096*bk  (each wave: tiles e0 = 16*wave, 16*(wave+8))
  for (int t = 0; t < 2; ++t) {
    const int e0 = (wave + t * 8) * 16;
    v8f acc = {};
    for (int k = 0; k < DIMC; k += 4) {
      v2f a, bf;
      a.x  = Wk[(e0 + nn) * DIMC + k + kh];
      a.y  = Wk[(e0 + nn) * DIMC + k + kh + 1];
      bf.x = S_lds[(k + kh) * 16 + nn];
      bf.y = S_lds[(k + kh + 1) * 16 + nn];
      acc = __builtin_amdgcn_wmma_f32_16x16x4_f32(false, a, false, bf,
                                                  (short)0, acc, false, false);
    }
#pragma unroll
    for (int r = 0; r < 8; ++r) {
      int e = e0 + rowoff + r;
      KS_lds[e * 16 + nn] = acc[r] + 4096.0f * bk[e];
    }
  }
  __syncthreads();

  // Phase 2: V = Wq^T * KS   -> vout[b*256 + d]
  for (int t = 0; t < 2; ++t) {
    const int d0 = (wave + t * 8) * 16;
    v8f acc = {};
    for (int k = 0; k < DIMC; k += 4) {
      v2f a, bf;
      a.x  = Wq[(k + kh) * DIMC + d0 + nn];         // A[M=d][K=e] = Wq[e][d]
      a.y  = Wq[(k + kh + 1) * DIMC + d0 + nn];
      bf.x = KS_lds[(k + kh) * 16 + nn];
      bf.y = KS_lds[(k + kh + 1) * 16 + nn];
      acc = __builtin_amdgcn_wmma_f32_16x16x4_f32(false, a, false, bf,
                                                  (short)0, acc, false, false);
    }
#pragma unroll
    for (int r = 0; r < 8; ++r) {
      int d = d0 + rowoff + r;
      vout[nn * DIMC + d] = acc[r];
    }
  }

  // Phase 3: c[b] = bq . KS[:,b]
  if (threadIdx.x < 16) {
    float c = 0.f;
    for (int e = 0; e < DIMC; ++e) c += bq[e] * KS_lds[e * 16 + threadIdx.x];
    cout[threadIdx.x] = c;
  }
}

// ------------------- Tensor Data Mover helpers (CDNA5 TDM, ISA ch.8) --------
__device__ inline uint32_t lds_offset_of(const void* p) {
  // Shared-aperture generic address: addr[31:0] == LDS byte offset.
  return (uint32_t)(uintptr_t)p;
}

// 2D tile load: tensor rows = tdim1 (stride stride0 elems), row length tdim0;
// tile = tile0 x tile1, f32 elements, packed contiguously into LDS at lds_off.
__device__ inline void tdm_load_2d_f32(uint32_t lds_off, const void* gsrc,
                                       uint32_t tdim0, uint32_t tdim1,
                                       uint32_t tile0, uint32_t tile1,
                                       uint64_t stride0 /*elements*/) {
  uint64_t ga = (uint64_t)(uintptr_t)gsrc;
  uint64_t s1 = stride0 * (uint64_t)tdim1;    // dim1 stride (unused: tile_dim2=0)
  v4u g0;
  g0.x = 1u;                                        // count=1, user descriptor
  g0.y = lds_off;                                   // lds_addr (bytes)
  g0.z = (uint32_t)ga;                              // global_addr[31:0]
  g0.w = ((uint32_t)(ga >> 32) & 0x01FFFFFFu) | (2u << 30);  // addr[56:32] | type=2
  v8i g1;
  g1[0] = (int)(2u << 16);                          // data_size=4B; no pad/iter/multicast
  g1[1] = (int)(tdim0 << 16);                       // tensor_dim0 -> bits[79:48]
  g1[2] = (int)((tdim0 >> 16) | (tdim1 << 16));     // tensor_dim1 -> bits[111:80]
  g1[3] = (int)((tdim1 >> 16) | (tile0 << 16));     // tile_dim0  -> bits[127:112]
  g1[4] = (int)(tile1 & 0xFFFFu);                   // tile_dim1; tile_dim2 = 0
  g1[5] = (int)(uint32_t)stride0;                   // tensor_dim0_stride[31:0]
  g1[6] = (int)(((uint32_t)(stride0 >> 32) & 0xFFFFu) |
                ((uint32_t)(s1 & 0xFFFFu) << 16));  // d0s[47:32] | d1s[15:0]
  g1[7] = (int)(uint32_t)(s1 >> 16);                // tensor_dim1_stride[47:16]
  v4i g2 = {0, 0, 0, 0};
  v4i g3 = {0, 0, 0, 0};
#if __clang_major__ >= 23
  v8i g4 = {0, 0, 0, 0, 0, 0, 0, 0};
  __builtin_amdgcn_tensor_load_to_lds(g0, g1, g2, g3, g4, 0);
#else
  __builtin_amdgcn_tensor_load_to_lds(g0, g1, g2, g3, 0);
#endif
}

// ------------------- kernel 3: dot + softmax + blend, TDM-staged LDS --------
__global__ __launch_bounds__(256) void k3_fuse(const float* __restrict__ rgb,
                                               const float* __restrict__ evt,
                                               const float* __restrict__ ws,
                                               float* __restrict__ out) {
  __shared__ float tileA[DIMC * TILE_T];   // [ch][tok]  64 KB
  __shared__ float tileB[DIMC * TILE_T];   // 64 KB
  __shared__ float vA[DIMC], vD[DIMC];
  __shared__ float redA[256], redB[256];
  __shared__ float wA[TILE_T], wB[TILE_T];

  const int b   = blockIdx.x >> 6;                  // 64 token-chunks per batch
  const int n0  = (blockIdx.x & 63) * TILE_T;
  const int tid = threadIdx.x;
  const size_t base = (size_t)b * DIMC * NTOK + n0;

  if (tid < 32) {                                   // wave 0 drives the TDM
    tdm_load_2d_f32(lds_offset_of(tileA), rgb + base, NTOK, DIMC, TILE_T, DIMC, NTOK);
    tdm_load_2d_f32(lds_offset_of(tileB), evt + base, NTOK, DIMC, TILE_T, DIMC, NTOK);
  }
  vA[tid] = ws[2 * BATCH * DIMC + b * DIMC + tid];               // V (branch a)
  vD[tid] = ws[2 * BATCH * DIMC + BATCH * DIMC + b * DIMC + tid];
  if (tid < 32) __builtin_amdgcn_s_wait_tensorcnt(0);
  __syncthreads();

  const int tok  = tid & (TILE_T - 1);
  const int part = tid >> 6;                        // 4 partials per token
  float pa = 0.f, pd = 0.f;
#pragma unroll 4
  for (int c = 0; c < 64; ++c) {
    int ch = part * 64 + c;
    pa += tileA[ch * TILE_T + tok] * vA[ch];
    pd += tileB[ch * TILE_T + tok] * vD[ch];
  }
  redA[tid] = pa;
  redB[tid] = pd;
  __syncthreads();

  if (tid < TILE_T) {
    float ca = ws[4 * BATCH * DIMC + b];
    float cd = ws[4 * BATCH * DIMC + BATCH + b];
    float va = (redA[tid] + redA[64 + tid] + redA[128 + tid] + redA[192 + tid] + ca) * 0.0625f;
    float vd = (redB[tid] + redB[64 + tid] + redB[128 + tid] + redB[192 + tid] + cd) * 0.0625f;
    float m  = fmaxf(va, vd);
    float ea = __expf(va - m), ed = __expf(vd - m);
    float inv = 1.0f / (ea + ed);
    wA[tid] = ea * inv;
    wB[tid] = ed * inv;
  }
  __syncthreads();

  const float wa = wA[tok], wb = wB[tok];
  for (int c = 0; c < 64; ++c) {
    int ch = part * 64 + c;
    out[base + (size_t)ch * NTOK + tok] =
        wa * tileA[ch * TILE_T + tok] + wb * tileB[ch * TILE_T + tok];
  }
}

// ---------------------------------------------------------------------------
extern "C" void kernel_launch(void* const* d_in, const int* in_sizes, int n_in,
                              void* d_out, int out_size, void* d_ws, size_t ws_size,
                              hipStream_t stream) {
  const float* rgb  = (const float*)d_in[0];
  const float* evt  = (const float*)d_in[1];
  const float* Wq_a = (const float*)d_in[2];
  const float* bq_a = (const float*)d_in[3];
  const float* Wk_a = (const float*)d_in[4];
  const float* bk_a = (const float*)d_in[5];
  const float* Wq_d = (const float*)d_in[6];
  const float* bq_d = (const float*)d_in[7];
  const float* Wk_d = (const float*)d_in[8];
  const float* bk_d = (const float*)d_in[9];
  float* out = (float*)d_out;
  float* ws  = (float*)d_ws;   // needs ~66 KB of scratch

  k1_sums<<<(2 * BATCH * DIMC) / 8, 256, 0, stream>>>(rgb, evt, ws);
  k2_proj_wmma<<<2, 256, 0, stream>>>(Wq_a, bq_a, Wk_a, bk_a,
                                      Wq_d, bq_d, Wk_d, bk_d, ws);
  k3_fuse<<<BATCH * (NTOK / TILE_T), 256, 0, stream>>>(rgb, evt, ws, out);
}